// AttentionDecoder_26568667693682
// MI455X (gfx1250) — compile-verified
//
#include <hip/hip_runtime.h>
#include <math.h>
#include <stdint.h>

typedef __attribute__((ext_vector_type(2))) float v2f;
typedef __attribute__((ext_vector_type(8))) float v8f;

#define HD 1024
#define LL 4096
#define VV 50257

// ---------------------------------------------------------------------------
// Wave-level fp32 WMMA GEMV: out[r] = dot(Mat[r, 0:K], vec[0:K]) + bias[r]
// Each wave owns a 16-row tile; K consumed 4 at a time via V_WMMA_F32_16X16X4_F32.
// A(16x4): lanes 0-15 hold (row, K=0..1), lanes 16-31 hold (row, K=2..3).
// B(4x16): replicated across columns -> per-lane {vec[k0+koff], vec[k0+koff+1]}.
//   (B layout: lanes 0-15 = K lower half, lanes 16-31 = K upper half, one K-row
//    per VGPR — matches both the C/D striping rule and the sparse-B tables.)
// D: lane 0 has y[0..7] in acc[0..7]; lane 16 has y[8..15].
// Vector operand staged into LDS via CDNA5 async copy (ASYNCcnt path).
// ---------------------------------------------------------------------------
__global__ __launch_bounds__(256) void wmma_gemv(
    const float* __restrict__ Mat, const float* __restrict__ vec,
    const float* __restrict__ bias, float* __restrict__ out,
    int Mrows, int K, int ntiles)
{
  __shared__ float sv[2048];                  // K <= 2048 for all uses
  const int t = threadIdx.x;

  // Async global->LDS stage of the shared vector (16B per thread per pass).
  for (int i = 4 * t; i < K; i += 1024) {
    uint32_t lds_addr = (uint32_t)(uintptr_t)(&sv[i]);   // addr[31:0] = LDS offset
    uint32_t goff     = (uint32_t)(i * 4);
    asm volatile("global_load_async_to_lds_b128 %0, %1, %2"
                 :: "v"(lds_addr), "v"(goff), "s"(vec) : "memory");
  }
  asm volatile("s_wait_asynccnt 0x0" ::: "memory");
  __syncthreads();

  const int wave = t >> 5;
  const int lane = t & 31;
  const int tile = blockIdx.x * 8 + wave;
  if (tile < ntiles) {
    const int m    = lane & 15;               // row within 16-row tile
    const int koff = (lane >> 4) << 1;        // 0 for lanes 0-15, 2 for 16-31
    int row  = tile * 16 + m;
    int lrow = row < Mrows ? row : Mrows - 1; // clamp loads in last tile
    const float* rp = Mat + (size_t)lrow * K + koff;

    v8f acc = {};
    for (int k0 = 0; k0 < K; k0 += 4) {
      if ((k0 & 255) == 0)
        __builtin_prefetch(rp + k0 + 512, 0, 1);   // global_prefetch_b8
      v2f A = *(const v2f*)(rp + k0);              // 8B stream per lane-row
      v2f B = { sv[k0 + koff], sv[k0 + koff + 1] }; // LDS broadcast
      acc = __builtin_amdgcn_wmma_f32_16x16x4_f32(
                false, A, false, B, (short)0, acc, false, false);
    }

    if (lane == 0 || lane == 16) {
      int base = tile * 16 + ((lane >> 4) << 3);
      #pragma unroll
      for (int q = 0; q < 8; ++q) {
        int r = base + q;
        if (r < Mrows) out[r] = acc[q] + (bias ? bias[r] : 0.0f);
      }
    }
  }
}

// ---------------------------------------------------------------------------
// softmax over 4096 scores (shift-invariant: the s_prev·Ws + align_b constant
// was legally dropped from e). Writes a[] to workspace and to d_out slot.
// ---------------------------------------------------------------------------
__global__ __launch_bounds__(1024) void softmax4096(
    const float* __restrict__ e, float* __restrict__ a_ws,
    float* __restrict__ a_out)
{
  __shared__ float red[1024];
  const int t = threadIdx.x;

  float mx = -INFINITY;
  for (int i = t; i < LL; i += 1024) mx = fmaxf(mx, e[i]);
  red[t] = mx; __syncthreads();
  for (int s = 512; s > 0; s >>= 1) {
    if (t < s) red[t] = fmaxf(red[t], red[t + s]);
    __syncthreads();
  }
  mx = red[0]; __syncthreads();

  float sum = 0.0f;
  for (int i = t; i < LL; i += 1024) {
    float v = expf(e[i] - mx);
    a_ws[i] = v;
    sum += v;
  }
  red[t] = sum; __syncthreads();
  for (int s = 512; s > 0; s >>= 1) {
    if (t < s) red[t] += red[t + s];
    __syncthreads();
  }
  float inv = 1.0f / red[0];

  for (int i = t; i < LL; i += 1024) {
    float av = a_ws[i] * inv;
    a_ws[i]  = av;
    a_out[i] = av;
  }
}

// ---------------------------------------------------------------------------
// c = a @ h_all, phase 1: deterministic row-chunk partials so the 16 MB h_all
// stream spreads over 128 workgroups (bandwidth, not latency, limited).
// Block b: col-block cb = b & 3 (256 cols), row chunk = b >> 2 (128 rows).
// ---------------------------------------------------------------------------
__global__ __launch_bounds__(256) void ctx_partial(
    const float* __restrict__ a, const float* __restrict__ h_all,
    float* __restrict__ part)
{
  const int cb    = blockIdx.x & 3;
  const int chunk = blockIdx.x >> 2;          // 0..31
  const int j     = cb * 256 + threadIdx.x;   // column 0..1023
  const int i0    = chunk * 128;

  const float* hp = h_all + (size_t)i0 * HD + j;
  float acc = 0.0f;
  #pragma unroll 4
  for (int i = 0; i < 128; ++i) {
    if ((i & 31) == 0) __builtin_prefetch(hp + (size_t)(i + 32) * HD, 0, 1);
    acc += a[i0 + i] * hp[(size_t)i * HD];
  }
  part[(size_t)chunk * HD + j] = acc;
}

// Phase 2: fold 32 partials per column; also fill xcat = [emb_W[y], c].
__global__ __launch_bounds__(256) void ctx_reduce_embed(
    const float* __restrict__ part, const float* __restrict__ emb_W,
    const int* __restrict__ yb, float* __restrict__ xcat)
{
  const int j = blockIdx.x * 256 + threadIdx.x;   // 0..1023
  float acc = 0.0f;
  #pragma unroll
  for (int c = 0; c < 32; ++c) acc += part[(size_t)c * HD + j];
  xcat[HD + j] = acc;
  xcat[j] = emb_W[(size_t)yb[0] * HD + j];
}

// ---------------------------------------------------------------------------
// GRU gate fusion: r,z,n -> h_new. Writes h_new to workspace + d_out slot.
// ---------------------------------------------------------------------------
__global__ __launch_bounds__(256) void gru_gates(
    const float* __restrict__ gi, const float* __restrict__ gh,
    const float* __restrict__ s_prev, float* __restrict__ h_ws,
    float* __restrict__ h_out)
{
  const int k = blockIdx.x * 256 + threadIdx.x;   // 0..1023
  float r = 1.0f / (1.0f + expf(-(gi[k]          + gh[k])));
  float z = 1.0f / (1.0f + expf(-(gi[HD + k]     + gh[HD + k])));
  float n = tanhf(gi[2 * HD + k] + r * gh[2 * HD + k]);
  float h = (1.0f - z) * n + z * s_prev[k];
  h_ws[k]  = h;
  h_out[k] = h;
}

// ---------------------------------------------------------------------------
// log_softmax over 50257 logits (resident in L2 after the big GEMV).
// ---------------------------------------------------------------------------
__global__ __launch_bounds__(1024) void logsoftmaxV(
    const float* __restrict__ lg, float* __restrict__ outp)
{
  __shared__ float red[1024];
  const int t = threadIdx.x;

  float mx = -INFINITY;
  for (int v = t; v < VV; v += 1024) mx = fmaxf(mx, lg[v]);
  red[t] = mx; __syncthreads();
  for (int s = 512; s > 0; s >>= 1) {
    if (t < s) red[t] = fmaxf(red[t], red[t + s]);
    __syncthreads();
  }
  mx = red[0]; __syncthreads();

  float sum = 0.0f;
  for (int v = t; v < VV; v += 1024) sum += expf(lg[v] - mx);
  red[t] = sum; __syncthreads();
  for (int s = 512; s > 0; s >>= 1) {
    if (t < s) red[t] += red[t + s];
    __syncthreads();
  }
  float off = mx + logf(red[0]);

  for (int v = t; v < VV; v += 1024) outp[v] = lg[v] - off;
}

// ---------------------------------------------------------------------------
extern "C" void kernel_launch(void* const* d_in, const int* in_sizes, int n_in,
                              void* d_out, int out_size, void* d_ws, size_t ws_size,
                              hipStream_t stream) {
  const int*   y_before = (const int*)  d_in[0];
  const float* s_before = (const float*)d_in[1];   // (1,1,H)
  const float* h_all    = (const float*)d_in[2];   // (L,H)
  const float* emb_W    = (const float*)d_in[3];   // (V,H)
  const float* align_W  = (const float*)d_in[4];   // (1,2H): first H = Wh
  // align_b, s_prev·Ws: constants across rows -> dropped (softmax invariant)
  const float* new_W    = (const float*)d_in[6];   // (H,2H)
  const float* new_b    = (const float*)d_in[7];
  const float* W_ih     = (const float*)d_in[8];   // (3H,H)
  const float* b_ih     = (const float*)d_in[9];
  const float* W_hh     = (const float*)d_in[10];  // (3H,H)
  const float* b_hh     = (const float*)d_in[11];
  const float* out_W    = (const float*)d_in[12];  // (V,H)
  const float* out_b    = (const float*)d_in[13];

  float* out = (float*)d_out;          // [logp(V) | h_new(H) | a(L)]
  float* ws  = (float*)d_ws;
  float* ws_e      = ws;               // 4096
  float* ws_a      = ws + 4096;        // 4096
  float* ws_xcat   = ws + 8192;        // 2048
  float* ws_x      = ws + 10240;       // 1024
  float* ws_gi     = ws + 11264;       // 3072
  float* ws_gh     = ws + 14336;       // 3072
  float* ws_h      = ws + 17408;       // 1024
  float* ws_logits = ws + 18432;       // 50257 (ends 68689)
  float* ws_part   = ws + 69632;       // 32*1024 ctx partials

  // 1) e = h_all @ Wh               (4096 rows, K=1024, 256 tiles)
  wmma_gemv<<<32, 256, 0, stream>>>(h_all, align_W, nullptr, ws_e, LL, HD, 256);
  // 2) a = softmax(e)
  softmax4096<<<1, 1024, 0, stream>>>(ws_e, ws_a, out + VV + HD);
  // 3) c = a @ h_all (128-block partials, deterministic) ; xcat = [emb, c]
  ctx_partial<<<128, 256, 0, stream>>>(ws_a, h_all, ws_part);
  ctx_reduce_embed<<<4, 256, 0, stream>>>(ws_part, emb_W, y_before, ws_xcat);
  // 4) x = new_W @ xcat + new_b     (1024 rows, K=2048, 64 tiles)
  wmma_gemv<<<8, 256, 0, stream>>>(new_W, ws_xcat, new_b, ws_x, HD, 2 * HD, 64);
  // 5) gi = W_ih @ x + b_ih         (3072 rows, K=1024, 192 tiles)
  wmma_gemv<<<24, 256, 0, stream>>>(W_ih, ws_x, b_ih, ws_gi, 3 * HD, HD, 192);
  // 6) gh = W_hh @ s_prev + b_hh
  wmma_gemv<<<24, 256, 0, stream>>>(W_hh, s_before, b_hh, ws_gh, 3 * HD, HD, 192);
  // 7) GRU gates -> h_new
  gru_gates<<<4, 256, 0, stream>>>(ws_gi, ws_gh, s_before, ws_h, out + VV);
  // 8) logits = out_W @ h_new + out_b   (50257 rows -> 3142 tiles, 393 blocks)
  wmma_gemv<<<393, 256, 0, stream>>>(out_W, ws_h, out_b, ws_logits, VV, HD, 3142);
  // 9) logp = log_softmax(logits)
  logsoftmaxV<<<1, 1024, 0, stream>>>(ws_logits, out);
}